// CrossLayerPoolLight_51170240364943
// MI455X (gfx1250) — compile-verified
//
#include <hip/hip_runtime.h>
#include <cstddef>

typedef __attribute__((ext_vector_type(2))) float v2f;
typedef __attribute__((ext_vector_type(8))) float v8f;

#define CH 64
#define KNN 16
#define LEAKY 0.1f

// ---------------------------------------------------------------------------
// Batched 2D transpose: in [B, rows, cols] -> out [B, cols, rows]
// grid: (cols/16, rows/16, B), block (16,16)
// ---------------------------------------------------------------------------
__global__ void __launch_bounds__(256)
transpose2d_kernel(const float* __restrict__ in, float* __restrict__ out,
                   int rows, int cols) {
  __shared__ float tile[16][17];
  const size_t base = (size_t)blockIdx.z * rows * cols;
  const int c0 = blockIdx.x * 16, r0 = blockIdx.y * 16;
  tile[threadIdx.y][threadIdx.x] =
      in[base + (size_t)(r0 + threadIdx.y) * cols + (c0 + threadIdx.x)];
  __syncthreads();
  out[base + (size_t)(c0 + threadIdx.y) * rows + (r0 + threadIdx.x)] =
      tile[threadIdx.x][threadIdx.y];
}

// ---------------------------------------------------------------------------
// Brute-force KNN (top-16 smallest squared distances), computed ONCE per
// direction and reused by every _cross call (indices depend only on pc1/pc2).
// xyzQ, xyzR: [B,3,N].  idx out: [B,N,16].
// One thread per query point; reference points tiled through LDS.
// Register-resident sorted insertion list (all indices static after unroll).
// ---------------------------------------------------------------------------
__global__ void __launch_bounds__(256)
knn_kernel(const float* __restrict__ xyzQ, const float* __restrict__ xyzR,
           int* __restrict__ idx, int N) {
  const int TILE = 256;
  __shared__ float sx[TILE], sy[TILE], sz[TILE];
  const int q = blockIdx.x * blockDim.x + threadIdx.x;  // global query in B*N
  const int b = q / N, n = q % N;
  const float* Q = xyzQ + (size_t)b * 3 * N;
  const float* R = xyzR + (size_t)b * 3 * N;
  const float qx = Q[n], qy = Q[N + n], qz = Q[2 * N + n];

  float best[KNN];
  int bidx[KNN];
#pragma unroll
  for (int s = 0; s < KNN; ++s) { best[s] = 3.4e38f; bidx[s] = 0; }

  for (int t0 = 0; t0 < N; t0 += TILE) {
    const int j = t0 + threadIdx.x;
    sx[threadIdx.x] = R[j];
    sy[threadIdx.x] = R[N + j];
    sz[threadIdx.x] = R[2 * N + j];
    __syncthreads();
    for (int t = 0; t < TILE; ++t) {
      const float dx = sx[t] - qx, dy = sy[t] - qy, dz = sz[t] - qz;
      const float d = dx * dx + dy * dy + dz * dz;
      if (d < best[KNN - 1]) {               // amortized: rare after warmup
        int rank = 0;
#pragma unroll
        for (int s = 0; s < KNN; ++s) rank += (best[s] <= d) ? 1 : 0;
#pragma unroll
        for (int s = KNN - 1; s >= 1; --s)
          if (s > rank) { best[s] = best[s - 1]; bidx[s] = bidx[s - 1]; }
#pragma unroll
        for (int s = 0; s < KNN; ++s)
          if (s == rank) { best[s] = d; bidx[s] = t0 + t; }
      }
    }
    __syncthreads();
  }
  int* o = idx + ((size_t)b * N + n) * KNN;
#pragma unroll
  for (int s = 0; s < KNN; ++s) o[s] = bidx[s];
}

// ---------------------------------------------------------------------------
// 1x1 conv as WMMA GEMM (f32, 16x16x4):  Out[b,n,o] = sum_c F[b,n,c]*W[o,c]+bias[o]
// F, Out: [B, N, 64] (NHWC).  W: [64,64] row-major (o-major).  One wave per
// 16(point) x 16(outch) D tile, K-loop over 64 channels in steps of 4.
// A frag (F tile, M=point,K=ch) and B frag (W^T tile, K=ch,N=outch) are both a
// single float2 at  base + (tile0+(lane&15))*64 + k + 2*(lane>>4).
// EXEC is all-1s: grid exactly covers the tile space (WMMA requirement).
// ---------------------------------------------------------------------------
__global__ void __launch_bounds__(256)
conv1x1_wmma_kernel(const float* __restrict__ F, const float* __restrict__ W,
                    const float* __restrict__ bias, float* __restrict__ Out,
                    int N) {
  const int ntiles = N >> 4;                       // 16-point tiles per batch
  const int wave = (blockIdx.x * blockDim.x + threadIdx.x) >> 5;
  const int lane = threadIdx.x & 31;
  const int otile = wave & 3;                      // 64 outch -> 4 tiles
  const int ntile = (wave >> 2) % ntiles;
  const int b = (wave >> 2) / ntiles;
  const int n0 = ntile << 4, o0 = otile << 4;
  const int lr = lane & 15, lh = lane >> 4;        // row in half, half select

  const float* Fp = F + (size_t)b * N * CH + (size_t)(n0 + lr) * CH + 2 * lh;
  const float* Wp = W + (size_t)(o0 + lr) * CH + 2 * lh;

  v8f acc = {};
#pragma unroll
  for (int k = 0; k < CH; k += 4) {
    const v2f a = *(const v2f*)(Fp + k);           // 8B aligned (k, 2*lh even)
    const v2f bb = *(const v2f*)(Wp + k);
    acc = __builtin_amdgcn_wmma_f32_16x16x4_f32(
        /*neg_a=*/false, a, /*neg_b=*/false, bb,
        /*c_mod=*/(short)0, acc, /*reuse_a=*/false, /*reuse_b=*/false);
  }

  const float bv = bias[o0 + lr];
  float* Op = Out + (size_t)b * N * CH + (size_t)(n0 + 8 * lh) * CH + o0 + lr;
#pragma unroll
  for (int j = 0; j < 8; ++j)                      // D VGPR j: rows j / j+8
    Op[(size_t)j * CH] = acc[j] + bv;
}

// ---------------------------------------------------------------------------
// Cross gather + pos-MLP + leaky-relu + max over K neighbors.
// out[b,n,c] = max_k leaky( p1[b,n,c] + p2[b,idx,c] + posW[c,:].(x2[idx]-x1[n])
//                           + posb[c] + bias[c] )
// block (64,4): threadIdx.x = channel (coalesced 256B gathers of p2 rows),
// threadIdx.y = point. xyz broadcast loads hit L2/WGP$.
// ---------------------------------------------------------------------------
__global__ void __launch_bounds__(256)
cross_pool_kernel(const float* __restrict__ xyz1, const float* __restrict__ xyz2,
                  const int* __restrict__ idx, const float* __restrict__ p1,
                  const float* __restrict__ p2, const float* __restrict__ posW,
                  const float* __restrict__ posb, const float* __restrict__ lbias,
                  float* __restrict__ out, int N) {
  const int c = threadIdx.x;                                     // 0..63
  const int pt = blockIdx.x * blockDim.y + threadIdx.y;          // in B*N
  const int b = pt / N, n = pt % N;
  const float* x1 = xyz1 + (size_t)b * 3 * N;
  const float* x2 = xyz2 + (size_t)b * 3 * N;
  const float qx = x1[n], qy = x1[N + n], qz = x1[2 * N + n];
  const float w0 = posW[c * 3 + 0], w1 = posW[c * 3 + 1], w2 = posW[c * 3 + 2];
  const float base = p1[((size_t)b * N + n) * CH + c] + posb[c] + lbias[c];
  const int* id = idx + ((size_t)b * N + n) * KNN;
  const float* p2b = p2 + (size_t)b * N * CH;

  float best = -3.4e38f;
#pragma unroll 4
  for (int k = 0; k < KNN; ++k) {
    const int j = id[k];
    const float dx = x2[j] - qx, dy = x2[N + j] - qy, dz = x2[2 * N + j] - qz;
    float v = base + p2b[(size_t)j * CH + c] + w0 * dx + w1 * dy + w2 * dz;
    v = v >= 0.0f ? v : LEAKY * v;
    best = fmaxf(best, v);
  }
  out[((size_t)b * N + n) * CH + c] = best;
}

// ---------------------------------------------------------------------------
// Host orchestration
// ---------------------------------------------------------------------------
extern "C" void kernel_launch(void* const* d_in, const int* in_sizes, int n_in,
                              void* d_out, int out_size, void* d_ws,
                              size_t ws_size, hipStream_t stream) {
  (void)in_sizes; (void)n_in; (void)out_size; (void)ws_size;
  const int B = 4, N = 4096;

  const float* pc1 = (const float*)d_in[0];   // [B,3,N]
  const float* pc2 = (const float*)d_in[1];   // [B,3,N]
  const float* feat1 = (const float*)d_in[2]; // [B,64,N]
  const float* feat2 = (const float*)d_in[3]; // [B,64,N]

  struct Layer { const float *posW, *posb, *c1W, *c1b, *c2W, *c2b, *bias; };
  Layer L[3];
  for (int l = 0; l < 3; ++l) {
    const int a = 4 + 7 * l;  // dict order: posW, posb, c1W, c1b, c2W, c2b, bias
    L[l] = {(const float*)d_in[a + 0], (const float*)d_in[a + 1],
            (const float*)d_in[a + 2], (const float*)d_in[a + 3],
            (const float*)d_in[a + 4], (const float*)d_in[a + 5],
            (const float*)d_in[a + 6]};
  }

  // Workspace carve-up (~34 MB)
  const size_t featElems = (size_t)B * N * CH;              // 1,048,576
  const size_t featBytes = featElems * sizeof(float);       // 4 MB
  const size_t idxBytes = (size_t)B * N * KNN * sizeof(int);// 1 MB
  char* w = (char*)d_ws;
  int* idx12 = (int*)w;  w += idxBytes;
  int* idx21 = (int*)w;  w += idxBytes;
  float* f1  = (float*)w; w += featBytes;   // current f1, NHWC
  float* f2  = (float*)w; w += featBytes;   // current f2, NHWC
  float* p1a = (float*)w; w += featBytes;
  float* p2a = (float*)w; w += featBytes;
  float* p1b = (float*)w; w += featBytes;
  float* p2b = (float*)w; w += featBytes;
  float* na  = (float*)w; w += featBytes;   // cross outputs (ping-pong)
  float* nb  = (float*)w; w += featBytes;

  const dim3 tBlk(16, 16);
  const dim3 tGridCN(N / 16, CH / 16, B);   // [B,C,N] -> [B,N,C]
  const dim3 tGridNC(CH / 16, N / 16, B);   // [B,N,C] -> [B,C,N]
  const int knnBlocks = (B * N) / 256;
  const int convBlocks = (B * (N / 16) * 4) / 8;  // 8 waves per 256-thr block
  const dim3 crossBlk(64, 4);
  const int crossBlocks = (B * N) / 4;

  // 1) Features to NHWC
  transpose2d_kernel<<<tGridCN, tBlk, 0, stream>>>(feat1, f1, CH, N);
  transpose2d_kernel<<<tGridCN, tBlk, 0, stream>>>(feat2, f2, CH, N);

  // 2) KNN once per direction (reused by all 5 cross calls)
  knn_kernel<<<knnBlocks, 256, 0, stream>>>(pc1, pc2, idx12, N);
  knn_kernel<<<knnBlocks, 256, 0, stream>>>(pc2, pc1, idx21, N);

  // 3) Two params1 layers (bidirectional cross)
  for (int l = 0; l < 2; ++l) {
    conv1x1_wmma_kernel<<<convBlocks, 256, 0, stream>>>(f1, L[l].c1W, L[l].c1b, p1a, N);
    conv1x1_wmma_kernel<<<convBlocks, 256, 0, stream>>>(f2, L[l].c2W, L[l].c2b, p2a, N);
    conv1x1_wmma_kernel<<<convBlocks, 256, 0, stream>>>(f2, L[l].c1W, L[l].c1b, p1b, N);
    conv1x1_wmma_kernel<<<convBlocks, 256, 0, stream>>>(f1, L[l].c2W, L[l].c2b, p2b, N);
    cross_pool_kernel<<<crossBlocks, crossBlk, 0, stream>>>(
        pc1, pc2, idx12, p1a, p2a, L[l].posW, L[l].posb, L[l].bias, na, N);
    cross_pool_kernel<<<crossBlocks, crossBlk, 0, stream>>>(
        pc2, pc1, idx21, p1b, p2b, L[l].posW, L[l].posb, L[l].bias, nb, N);
    float* t;
    t = f1; f1 = na; na = t;   // ping-pong
    t = f2; f2 = nb; nb = t;
  }

  // 4) params2 layer: ff = cross(pc1, pc2, conv1(f1), conv2(f2))
  conv1x1_wmma_kernel<<<convBlocks, 256, 0, stream>>>(f1, L[2].c1W, L[2].c1b, p1a, N);
  conv1x1_wmma_kernel<<<convBlocks, 256, 0, stream>>>(f2, L[2].c2W, L[2].c2b, p2a, N);
  cross_pool_kernel<<<crossBlocks, crossBlk, 0, stream>>>(
      pc1, pc2, idx12, p1a, p2a, L[2].posW, L[2].posb, L[2].bias, na, N);

  // 5) Outputs back to [B,C,N]: (f1, f2, ff) concatenated
  float* out = (float*)d_out;
  transpose2d_kernel<<<tGridNC, tBlk, 0, stream>>>(f1, out, N, CH);
  transpose2d_kernel<<<tGridNC, tBlk, 0, stream>>>(f2, out + featElems, N, CH);
  transpose2d_kernel<<<tGridNC, tBlk, 0, stream>>>(na, out + 2 * featElems, N, CH);
}